// _PointNet2SetAbstractionSSG_8246337208756
// MI455X (gfx1250) — compile-verified
//
#include <hip/hip_runtime.h>
#include <hip/hip_bf16.h>

typedef float f32x2 __attribute__((ext_vector_type(2)));
typedef float f32x8 __attribute__((ext_vector_type(8)));

#define B_BATCH   8
#define N_POINTS  8192
#define N_SAMP    2048
#define K_GROUP   32
#define F_IN      64
#define M_ROWS    (B_BATCH * N_SAMP * K_GROUP)   // 524288
#define RADIUS2   (0.2f * 0.2f)
#define BN_EPS    1e-5f

// ---------------------------------------------------------------------------
// 1. Furthest point sampling: one 1024-thread block per batch, points in regs.
// ---------------------------------------------------------------------------
#define FPS_T 1024
#define FPS_PPT (N_POINTS / FPS_T)   // 8

__global__ __launch_bounds__(FPS_T) void fps_kernel(
    const float* __restrict__ xyz, int* __restrict__ cidx,
    float* __restrict__ centroids /* d_out head: [B][2048][3] */) {
  const int b = blockIdx.x;
  const int t = threadIdx.x;
  const float* xb = xyz + (size_t)b * N_POINTS * 3;

  float px[FPS_PPT], py[FPS_PPT], pz[FPS_PPT], dist[FPS_PPT];
#pragma unroll
  for (int j = 0; j < FPS_PPT; ++j) {
    int i = t + j * FPS_T;
    px[j] = xb[i * 3 + 0];
    py[j] = xb[i * 3 + 1];
    pz[j] = xb[i * 3 + 2];
    dist[j] = 1e10f;
  }

  __shared__ float s_cur[3];
  __shared__ float s_wval[32];
  __shared__ int   s_widx[32];
  __shared__ int   s_last;

  if (t == 0) {
    cidx[b * N_SAMP] = 0;
    s_cur[0] = px[0]; s_cur[1] = py[0]; s_cur[2] = pz[0];
    float* c = centroids + ((size_t)b * N_SAMP) * 3;
    c[0] = px[0]; c[1] = py[0]; c[2] = pz[0];
  }
  __syncthreads();

  for (int s = 1; s < N_SAMP; ++s) {
    const float cx = s_cur[0], cy = s_cur[1], cz = s_cur[2];
    float best = -1.0f; int bidx = 0x7fffffff;
#pragma unroll
    for (int j = 0; j < FPS_PPT; ++j) {
      float dx = px[j] - cx, dy = py[j] - cy, dz = pz[j] - cz;
      float dd = dx * dx + dy * dy + dz * dz;
      dist[j] = fminf(dist[j], dd);
      int i = t + j * FPS_T;
      if (dist[j] > best || (dist[j] == best && i < bidx)) { best = dist[j]; bidx = i; }
    }
    // wave32 argmax (tie -> lower index)
#pragma unroll
    for (int off = 16; off > 0; off >>= 1) {
      float ov = __shfl_xor(best, off, 32);
      int   oi = __shfl_xor(bidx, off, 32);
      if (ov > best || (ov == best && oi < bidx)) { best = ov; bidx = oi; }
    }
    const int wave = t >> 5;
    if ((t & 31) == 0) { s_wval[wave] = best; s_widx[wave] = bidx; }
    __syncthreads();
    if (t < 32) {
      float v = s_wval[t]; int ix = s_widx[t];
#pragma unroll
      for (int off = 16; off > 0; off >>= 1) {
        float ov = __shfl_xor(v, off, 32);
        int   oi = __shfl_xor(ix, off, 32);
        if (ov > v || (ov == v && oi < ix)) { v = ov; ix = oi; }
      }
      if (t == 0) s_last = ix;
    }
    __syncthreads();
    const int last = s_last;
    if ((last & (FPS_T - 1)) == t) {   // owner thread broadcasts
      int j = last >> 10;
      s_cur[0] = px[j]; s_cur[1] = py[j]; s_cur[2] = pz[j];
      cidx[b * N_SAMP + s] = last;
      float* c = centroids + ((size_t)b * N_SAMP + s) * 3;
      c[0] = px[j]; c[1] = py[j]; c[2] = pz[j];
    }
    __syncthreads();
  }
}

// ---------------------------------------------------------------------------
// 2. Ball query: one wave per centroid; first K_GROUP ascending indices in r².
// ---------------------------------------------------------------------------
__global__ void ballq_kernel(const float* __restrict__ xyz,
                             const float* __restrict__ centroids,
                             int* __restrict__ gidx) {
  const int gid = (int)((blockIdx.x * blockDim.x + threadIdx.x) >> 5);
  const int lane = threadIdx.x & 31;
  if (gid >= B_BATCH * N_SAMP) return;
  const int b = gid >> 11;
  const float* xb = xyz + (size_t)b * N_POINTS * 3;
  const float* c = centroids + (size_t)gid * 3;
  const float cx = c[0], cy = c[1], cz = c[2];
  int* out = gidx + (size_t)gid * K_GROUP;

  int cnt = 0;
  for (int base = 0; base < N_POINTS && cnt < K_GROUP; base += 32) {
    const int i = base + lane;
    float dx = xb[i * 3 + 0] - cx;
    float dy = xb[i * 3 + 1] - cy;
    float dz = xb[i * 3 + 2] - cz;
    bool in = (dx * dx + dy * dy + dz * dz) <= RADIUS2;
    unsigned mask = (unsigned)__ballot(in);
    int pos = cnt + __popc(mask & ((1u << lane) - 1u));
    if (in && pos < K_GROUP) out[pos] = i;
    cnt += __popc(mask);
  }
  if (lane == 0) {
    if (cnt == 0) {
      for (int k = 0; k < K_GROUP; ++k) out[k] = 0;
    } else if (cnt < K_GROUP) {
      int f = out[0];   // same-wave store->load kept in order per ISA
      for (int k = cnt; k < K_GROUP; ++k) out[k] = f;
    }
  }
}

// ---------------------------------------------------------------------------
// 3. Gather + concat into x0 [M_ROWS][68]  (3 centered xyz | 64 feat | 1 pad)
// ---------------------------------------------------------------------------
#define CIN0_PAD 68
__global__ void gather_kernel(const float* __restrict__ xyz,
                              const float* __restrict__ feat,
                              const float* __restrict__ centroids,
                              const int* __restrict__ gidx,
                              float* __restrict__ x0) {
  const size_t total = (size_t)M_ROWS * CIN0_PAD;
  const size_t stride = (size_t)gridDim.x * blockDim.x;
  for (size_t i = (size_t)blockIdx.x * blockDim.x + threadIdx.x; i < total; i += stride) {
    const size_t row = i / CIN0_PAD;
    const int c = (int)(i % CIN0_PAD);
    const int g = (int)(row >> 5);
    const int b = g >> 11;
    const int idx = gidx[row];   // gidx flat index == row
    float v;
    if (c < 3) {
      v = xyz[((size_t)b * N_POINTS + idx) * 3 + c] - centroids[(size_t)g * 3 + c];
    } else if (c < 3 + F_IN) {
      v = feat[((size_t)b * N_POINTS + idx) * F_IN + (c - 3)];
    } else {
      v = 0.0f;
    }
    x0[i] = v;
  }
}

// ---------------------------------------------------------------------------
// 4. Weight prep: W[cout][cin] -> Wt[cinpad][cout], zero-padded K rows.
// ---------------------------------------------------------------------------
__global__ void wprep_kernel(const float* __restrict__ w, float* __restrict__ wt,
                             int cout, int cin, int cinpad) {
  const int i = blockIdx.x * blockDim.x + threadIdx.x;
  if (i >= cinpad * cout) return;
  const int k = i / cout, n = i % cout;
  wt[i] = (k < cin) ? w[n * cin + k] : 0.0f;
}

// ---------------------------------------------------------------------------
// 5. GEMM on the f32 WMMA path: Y[M][COUT] = act(A)[M][CIN] * Wt[CIN][COUT]+b
//    act = fused previous-layer BN+ReLU (per-channel scale/shift) when BN.
//    One wave owns a 16-row M tile and all COUT/16 N tiles; Wt staged in LDS.
// ---------------------------------------------------------------------------
template <int CIN, int COUT, bool BN>
__global__ __launch_bounds__(256) void gemm_kernel(
    const float* __restrict__ A, const float* __restrict__ Wt,
    const float* __restrict__ bias, const float* __restrict__ bn_scale,
    const float* __restrict__ bn_shift, float* __restrict__ Y) {
  __shared__ float sW[CIN * COUT];
  __shared__ float sB[COUT];
  __shared__ float sS[CIN];
  __shared__ float sH[CIN];
  for (int i = threadIdx.x; i < CIN * COUT; i += blockDim.x) sW[i] = Wt[i];
  for (int i = threadIdx.x; i < COUT; i += blockDim.x) sB[i] = bias[i];
  if (BN) {
    for (int i = threadIdx.x; i < CIN; i += blockDim.x) {
      sS[i] = bn_scale[i];
      sH[i] = bn_shift[i];
    }
  }
  __syncthreads();

  const int wave = threadIdx.x >> 5;
  const int lane = threadIdx.x & 31;
  const int nloc = lane & 15;
  const int hi   = lane >> 4;                 // selects K+2 half of the frag
  const long mtile = (long)blockIdx.x * (blockDim.x >> 5) + wave;

  constexpr int NT = COUT / 16;
  f32x8 acc[NT];
#pragma unroll
  for (int nt = 0; nt < NT; ++nt) {
    const float bv = sB[nt * 16 + nloc];      // D column == bias channel
#pragma unroll
    for (int r = 0; r < 8; ++r) acc[nt][r] = bv;
  }

  const long arow = mtile * 16 + nloc;        // A: lanes 0-15 / 16-31 both M=0..15
  const float* __restrict__ ap = A + arow * CIN;

  for (int k = 0; k < CIN; k += 4) {
    const int kb = k + 2 * hi;
    float a0 = ap[kb];
    float a1 = ap[kb + 1];
    if (BN) {
      a0 = fmaxf(a0 * sS[kb] + sH[kb], 0.0f);
      a1 = fmaxf(a1 * sS[kb + 1] + sH[kb + 1], 0.0f);
    }
    f32x2 afrag; afrag.x = a0; afrag.y = a1;
#pragma unroll
    for (int nt = 0; nt < NT; ++nt) {
      f32x2 bfrag;
      bfrag.x = sW[kb * COUT + nt * 16 + nloc];
      bfrag.y = sW[(kb + 1) * COUT + nt * 16 + nloc];
      acc[nt] = __builtin_amdgcn_wmma_f32_16x16x4_f32(
          false, afrag, false, bfrag, (short)0, acc[nt], false, false);
    }
  }

  const long rbase = mtile * 16 + hi * 8;     // D: VGPR r -> rows r / r+8
#pragma unroll
  for (int nt = 0; nt < NT; ++nt) {
#pragma unroll
    for (int r = 0; r < 8; ++r) {
      Y[(rbase + r) * COUT + nt * 16 + nloc] = acc[nt][r];
    }
  }
}

// ---------------------------------------------------------------------------
// 6. Per-channel sum / sumsq for BN stats.
// ---------------------------------------------------------------------------
template <int COUT>
__global__ __launch_bounds__(256) void stats_kernel(
    const float* __restrict__ Y, float* __restrict__ gsum,
    float* __restrict__ gsumsq) {
  const int c = threadIdx.x & (COUT - 1);
  constexpr int RPB = 256 / COUT;
  long row = (long)blockIdx.x * RPB + (threadIdx.x / COUT);
  const long rstride = (long)gridDim.x * RPB;
  float s = 0.0f, sq = 0.0f;
  for (; row < M_ROWS; row += rstride) {
    float v = Y[row * COUT + c];
    s += v;
    sq += v * v;
  }
  __shared__ float ss[256], ssq[256];
  ss[threadIdx.x] = s;
  ssq[threadIdx.x] = sq;
  __syncthreads();
  if (threadIdx.x < COUT) {
    float ts = 0.0f, tsq = 0.0f;
#pragma unroll
    for (int j = 0; j < RPB; ++j) {
      ts += ss[c + j * COUT];
      tsq += ssq[c + j * COUT];
    }
    atomicAdd(&gsum[c], ts);
    atomicAdd(&gsumsq[c], tsq);
  }
}

__global__ void bnfin_kernel(const float* __restrict__ gsum,
                             const float* __restrict__ gsumsq,
                             const float* __restrict__ g,
                             const float* __restrict__ beta,
                             float* __restrict__ scale,
                             float* __restrict__ shift, int cout) {
  const int c = threadIdx.x;
  if (c >= cout) return;
  const float inv = 1.0f / (float)M_ROWS;
  const float mean = gsum[c] * inv;
  const float var = gsumsq[c] * inv - mean * mean;
  const float a = g[c] * rsqrtf(var + BN_EPS);
  scale[c] = a;
  shift[c] = beta[c] - mean * a;
}

// ---------------------------------------------------------------------------
// 7. Final BN+ReLU + max over the K_GROUP dim -> d_out tail [B][2048][128]
// ---------------------------------------------------------------------------
__global__ __launch_bounds__(128) void maxout_kernel(
    const float* __restrict__ Y2, const float* __restrict__ scale,
    const float* __restrict__ shift, float* __restrict__ out) {
  const int g = blockIdx.x;        // 16384 groups
  const int c = threadIdx.x;       // 128 channels
  const float a = scale[c], b = shift[c];
  const float* base = Y2 + (size_t)g * K_GROUP * 128 + c;
  float m = 0.0f;                  // ReLU output is >= 0
#pragma unroll
  for (int kk = 0; kk < K_GROUP; ++kk) {
    m = fmaxf(m, fmaxf(base[kk * 128] * a + b, 0.0f));
  }
  out[(size_t)g * 128 + c] = m;
}

// ---------------------------------------------------------------------------
// Launch
// ---------------------------------------------------------------------------
extern "C" void kernel_launch(void* const* d_in, const int* in_sizes, int n_in,
                              void* d_out, int out_size, void* d_ws, size_t ws_size,
                              hipStream_t stream) {
  const float* xyz  = (const float*)d_in[0];
  const float* feat = (const float*)d_in[1];
  const float* w0 = (const float*)d_in[2];
  const float* b0 = (const float*)d_in[3];
  const float* g0 = (const float*)d_in[4];
  const float* be0 = (const float*)d_in[5];
  const float* w1 = (const float*)d_in[6];
  const float* b1 = (const float*)d_in[7];
  const float* g1 = (const float*)d_in[8];
  const float* be1 = (const float*)d_in[9];
  const float* w2 = (const float*)d_in[10];
  const float* b2 = (const float*)d_in[11];
  const float* g2 = (const float*)d_in[12];
  const float* be2 = (const float*)d_in[13];

  float* out = (float*)d_out;                 // [0,49152): centroids
  float* out_feat = out + B_BATCH * N_SAMP * 3;

  char* ws = (char*)d_ws;
  size_t off = 0;
  auto take = [&](size_t bytes) { char* p = ws + off; off += (bytes + 255) & ~(size_t)255; return p; };
  int*   cidx = (int*)take((size_t)B_BATCH * N_SAMP * 4);
  int*   gidx = (int*)take((size_t)B_BATCH * N_SAMP * K_GROUP * 4);
  float* wt0 = (float*)take((size_t)CIN0_PAD * 64 * 4);
  float* wt1 = (float*)take((size_t)64 * 64 * 4);
  float* wt2 = (float*)take((size_t)64 * 128 * 4);
  float* st0 = (float*)take(4 * 128 * 4);     // sum | sumsq | scale | shift
  float* st1 = (float*)take(4 * 128 * 4);
  float* st2 = (float*)take(4 * 128 * 4);
  float* x0 = (float*)take((size_t)M_ROWS * CIN0_PAD * 4);   // reused for y1
  float* y0 = (float*)take((size_t)M_ROWS * 64 * 4);
  float* y2 = (float*)take((size_t)M_ROWS * 128 * 4);
  float* y1 = x0;                              // x0 dead after layer 0

  // 1. FPS (centroids written straight into d_out)
  fps_kernel<<<B_BATCH, FPS_T, 0, stream>>>(xyz, cidx, out);

  // 2. weight transpose/pad (tiny)
  wprep_kernel<<<(CIN0_PAD * 64 + 255) / 256, 256, 0, stream>>>(w0, wt0, 64, 67, CIN0_PAD);
  wprep_kernel<<<(64 * 64 + 255) / 256, 256, 0, stream>>>(w1, wt1, 64, 64, 64);
  wprep_kernel<<<(64 * 128 + 255) / 256, 256, 0, stream>>>(w2, wt2, 128, 64, 64);

  // 3. ball query (one wave per centroid) + gather/concat
  ballq_kernel<<<(B_BATCH * N_SAMP * 32 + 255) / 256, 256, 0, stream>>>(xyz, out, gidx);
  gather_kernel<<<8192, 256, 0, stream>>>(xyz, feat, out, gidx, x0);

  const int gemm_blocks = (M_ROWS / 16) / 8;   // 8 waves (M tiles) per block

  // 4. layer 0: x0(68) -> y0(64)
  gemm_kernel<CIN0_PAD, 64, false><<<gemm_blocks, 256, 0, stream>>>(
      x0, wt0, b0, nullptr, nullptr, y0);
  hipMemsetAsync(st0, 0, 2 * 128 * 4, stream);
  stats_kernel<64><<<1024, 256, 0, stream>>>(y0, st0, st0 + 128);
  bnfin_kernel<<<1, 128, 0, stream>>>(st0, st0 + 128, g0, be0, st0 + 256, st0 + 384, 64);

  // 5. layer 1: BN0+ReLU fused into A load; y0(64) -> y1(64)
  gemm_kernel<64, 64, true><<<gemm_blocks, 256, 0, stream>>>(
      y0, wt1, b1, st0 + 256, st0 + 384, y1);
  hipMemsetAsync(st1, 0, 2 * 128 * 4, stream);
  stats_kernel<64><<<1024, 256, 0, stream>>>(y1, st1, st1 + 128);
  bnfin_kernel<<<1, 128, 0, stream>>>(st1, st1 + 128, g1, be1, st1 + 256, st1 + 384, 64);

  // 6. layer 2: BN1+ReLU fused; y1(64) -> y2(128)
  gemm_kernel<64, 128, true><<<gemm_blocks, 256, 0, stream>>>(
      y1, wt2, b2, st1 + 256, st1 + 384, y2);
  hipMemsetAsync(st2, 0, 2 * 128 * 4, stream);
  stats_kernel<128><<<1024, 256, 0, stream>>>(y2, st2, st2 + 128);
  bnfin_kernel<<<1, 128, 0, stream>>>(st2, st2 + 128, g2, be2, st2 + 256, st2 + 384, 128);

  // 7. BN2+ReLU + max over group -> output features
  maxout_kernel<<<B_BATCH * N_SAMP, 128, 0, stream>>>(y2, st2 + 256, st2 + 384, out_feat);
}